// RNNModel_52853867544622
// MI455X (gfx1250) — compile-verified
//
#include <hip/hip_runtime.h>

// Problem constants (match reference)
#define B_SZ 256
#define T_SZ 2048
#define DIN  64
#define H    128
#define DOUT 2
#define BT   16      // batch tile per workgroup
#define XPAD 68      // x tile row stride   (== 4 mod 64 banks -> conflict-free)
#define HPAD 132     // h-state row stride  (== 4 mod 64 banks -> conflict-free)

typedef __attribute__((ext_vector_type(2))) float v2f;
typedef __attribute__((ext_vector_type(8))) float v8f;

// All-LDS working set: 4 weight matrices pre-packed into WMMA B-operand
// layout, fused biases, double-buffered x tile and hidden states.
struct Smem {
  float Wih0p[8][16][32][2];   // 32 KB  (K=64  -> 16 K-blocks of 4)
  float Whh0p[8][32][32][2];   // 64 KB  (K=128 -> 32 K-blocks)
  float Wih1p[8][32][32][2];   // 64 KB
  float Whh1p[8][32][32][2];   // 64 KB
  float bias0[H];
  float bias1[H];
  float xs [2][BT][XPAD];
  float h1s[2][BT][HPAD];
  float h2s[2][BT][HPAD];
};                             // ~266.5 KB  (< 320 KB WGP LDS)

__device__ __forceinline__ v8f wmma4(v2f a, v2f b, v8f c) {
  // V_WMMA_F32_16X16X4_F32 : D(16x16,f32) = A(16x4,f32) x B(4x16,f32) + C
  return __builtin_amdgcn_wmma_f32_16x16x4_f32(
      /*neg_a=*/false, a, /*neg_b=*/false, b,
      /*c_mod=*/(short)0, c, /*reuse_a=*/false, /*reuse_b=*/false);
}

// Dual-accumulator K-panel MACs: breaks the dependent C->D chain in two and
// lets the scheduler keep two WMMAs in flight per wave.
//   Arow : &row[ko]  (this lane's A pair base, 8B aligned)
//   Bpan : (v2f*)panel + lane  (packed B panel, conflict-free b64 reads)
template <int KB>
__device__ __forceinline__ void mma_panel(const float* Arow, const v2f* Bpan,
                                          v8f& acc0, v8f& acc1) {
  const v2f* A = (const v2f*)__builtin_assume_aligned(Arow, 8);
#pragma unroll
  for (int kb = 0; kb < KB; kb += 2) {
    acc0 = wmma4(A[kb * 2],       Bpan[kb * 32],       acc0);
    acc1 = wmma4(A[(kb + 1) * 2], Bpan[(kb + 1) * 32], acc1);
  }
}

// Pack row-major W[128][K] into P[nt(8)][kb(K/4)][lane(32)][2]: each lane's
// v2f B-operand is one contiguous, bank-conflict-free ds_load_b64.
// B[k][n] = W[n][k]; lane<16 covers K = 4*kb+{0,1}, lane>=16 covers 4*kb+{2,3}.
__device__ __forceinline__ void pack_panels(const float* __restrict__ W,
                                            float* __restrict__ P,
                                            int K, int tid) {
  const int KB = K >> 2;
  const int total = 8 * KB * 32 * 2;
  for (int i = tid; i < total; i += 256) {
    const int j   = i & 1;
    const int l   = (i >> 1) & 31;
    const int blk = i >> 6;          // nt*KB + kb
    const int kb  = blk % KB;
    const int nt  = blk / KB;
    const int n   = nt * 16 + (l & 15);
    const int k   = kb * 4 + j + ((l >> 4) << 1);
    P[i] = W[n * K + k];
  }
}

__global__ void __launch_bounds__(256, 1)
rnn2_fused(const float* __restrict__ x,
           const float* __restrict__ Wih0, const float* __restrict__ Whh0,
           const float* __restrict__ bih0, const float* __restrict__ bhh0,
           const float* __restrict__ Wih1, const float* __restrict__ Whh1,
           const float* __restrict__ bih1, const float* __restrict__ bhh1,
           const float* __restrict__ fcw,  const float* __restrict__ fcb,
           float* __restrict__ out)
{
  extern __shared__ char smem_raw[];
  Smem& S = *reinterpret_cast<Smem*>(smem_raw);

  const int tid  = threadIdx.x;
  const int lane = tid & 31;
  const int wv   = tid >> 5;            // wave id 0..7 -> owns N-tile wv*16
  const int b0   = blockIdx.x * BT;     // batch tile origin

  // ---- one-time: pack weights + fuse biases into LDS ----
  pack_panels(Wih0, &S.Wih0p[0][0][0][0], DIN, tid);
  pack_panels(Whh0, &S.Whh0p[0][0][0][0], H,   tid);
  pack_panels(Wih1, &S.Wih1p[0][0][0][0], H,   tid);
  pack_panels(Whh1, &S.Whh1p[0][0][0][0], H,   tid);
  if (tid < H) {
    S.bias0[tid] = bih0[tid] + bhh0[tid];
    S.bias1[tid] = bih1[tid] + bhh1[tid];
  }
  for (int i = tid; i < BT * HPAD; i += 256) {   // h(t=0) = 0
    S.h1s[0][i / HPAD][i % HPAD] = 0.0f;
    S.h2s[0][i / HPAD][i % HPAD] = 0.0f;
  }
  const int r = tid >> 4, c = (tid & 15) << 2;   // x-tile load geometry
  {                                              // x tile for t = 0
    const float4 v = *reinterpret_cast<const float4*>(
        x + ((size_t)(b0 + r) * T_SZ + 0) * DIN + c);
    *reinterpret_cast<float4*>(&S.xs[0][r][c]) = v;
  }
  __syncthreads();

  // WMMA lane geometry (wave32)
  const int m  = lane & 15;             // A row (batch) this lane feeds
  const int ko = (lane >> 4) << 1;      // K sub-offset within 4-block: 0 or 2
  const int mm = (lane >> 4) << 3;      // C/D: +j gives output row (batch)
  const int nn = (wv << 4) + m;         // C/D output column (hidden unit)

  const v2f* Bih0p = (const v2f*)&S.Wih0p[wv][0][0][0] + lane;
  const v2f* Bhh0p = (const v2f*)&S.Whh0p[wv][0][0][0] + lane;
  const v2f* Bih1p = (const v2f*)&S.Wih1p[wv][0][0][0] + lane;
  const v2f* Bhh1p = (const v2f*)&S.Whh1p[wv][0][0][0] + lane;

  // loop-invariant fused biases -> registers (applied in the ReLU epilogue,
  // so accumulators start at 0 and the first WMMA gets SRC2 = inline 0)
  const float bv0 = S.bias0[nn];
  const float bv1 = S.bias1[nn];

  const v8f vzero = {0, 0, 0, 0, 0, 0, 0, 0};

  for (int t = 0; t < T_SZ; ++t) {
    const int cur = t & 1, nxt = cur ^ 1;

    // prefetch next x tile into registers; latency hidden under WMMA work
    const int tn = (t + 1 < T_SZ) ? (t + 1) : (T_SZ - 1);
    const float4 xr = *reinterpret_cast<const float4*>(
        x + ((size_t)(b0 + r) * T_SZ + tn) * DIN + c);

    // ---- layer 0: h1' = relu(x_t Wih0^T + h1 Whh0^T + bias0) ----
    v8f a0 = vzero, a1 = vzero;
    mma_panel<16>(&S.xs [cur][m][ko], Bih0p, a0, a1);
    mma_panel<32>(&S.h1s[cur][m][ko], Bhh0p, a0, a1);
#pragma unroll
    for (int j = 0; j < 8; ++j)
      S.h1s[nxt][mm + j][nn] = fmaxf(a0[j] + a1[j] + bv0, 0.0f);

    // ---- layer 1 partial: h2 Whh1^T  (independent of h1' -> pre-barrier,
    //      absorbs barrier slack: 80 of 112 WMMAs now before the sync) ----
    v8f c0 = vzero, c1 = vzero;
    mma_panel<32>(&S.h2s[cur][m][ko], Bhh1p, c0, c1);

    // stage prefetched x tile (read next iteration, after 2nd barrier)
    *reinterpret_cast<float4*>(&S.xs[nxt][r][c]) = xr;

    __syncthreads();   // h1' + staged x visible to all waves

    // ---- layer 1 rest: += h1' Wih1^T ; h2' = relu(. + bias1) ----
    mma_panel<32>(&S.h1s[nxt][m][ko], Bih1p, c0, c1);
#pragma unroll
    for (int j = 0; j < 8; ++j)
      S.h2s[nxt][mm + j][nn] = fmaxf(c0[j] + c1[j] + bv1, 0.0f);

    __syncthreads();   // h2' visible; safe to flip buffers
  }

  // ---- fused FC head: out[b,:] = h2_last[b,:] @ fcw^T + fcb ----
  const int fin = ((T_SZ - 1) & 1) ^ 1;   // buffer holding h2(T-1)
  if (tid < BT * DOUT) {
    const int mi = tid >> 1, o = tid & 1;
    float s = fcb[o];
#pragma unroll 8
    for (int k = 0; k < H; ++k) s += S.h2s[fin][mi][k] * fcw[o * H + k];
    out[(size_t)(b0 + mi) * DOUT + o] = s;
  }
}

extern "C" void kernel_launch(void* const* d_in, const int* in_sizes, int n_in,
                              void* d_out, int out_size, void* d_ws, size_t ws_size,
                              hipStream_t stream) {
  const float* x    = (const float*)d_in[0];
  const float* Wih0 = (const float*)d_in[1];
  const float* Whh0 = (const float*)d_in[2];
  const float* bih0 = (const float*)d_in[3];
  const float* bhh0 = (const float*)d_in[4];
  const float* Wih1 = (const float*)d_in[5];
  const float* Whh1 = (const float*)d_in[6];
  const float* bih1 = (const float*)d_in[7];
  const float* bhh1 = (const float*)d_in[8];
  const float* fcw  = (const float*)d_in[9];
  const float* fcb  = (const float*)d_in[10];
  float* out = (float*)d_out;

  static_assert(sizeof(Smem) <= 320 * 1024, "exceeds WGP LDS");
  (void)hipFuncSetAttribute((const void*)rnn2_fused,
                            hipFuncAttributeMaxDynamicSharedMemorySize,
                            (int)sizeof(Smem));

  rnn2_fused<<<B_SZ / BT, 256, sizeof(Smem), stream>>>(
      x, Wih0, Whh0, bih0, bhh0, Wih1, Whh1, bih1, bhh1, fcw, fcb, out);
}